// QuantumLayer_54125177865174
// MI455X (gfx1250) — compile-verified
//
#include <hip/hip_runtime.h>

typedef __attribute__((ext_vector_type(2))) float v2f;
typedef __attribute__((ext_vector_type(8))) float v8f;

#define NQ 10
#define NL 4
#define NSTATES 1024
#define WAVES_PER_BLOCK 4
#define THREADS (WAVES_PER_BLOCK * 32)

// LDS budget (bytes):
//   u2     : NL*NQ*8 floats          = 1280
//   UA     : 2 * NL*256 floats       = 8192
//   UB     : 2 * NL*256 floats       = 8192
//   UC     : 2 * NL*16  floats       = 512
//   permPk : NL*512 uint32           = 8192
//   state  : WAVES * 1024 float2     = 32768   (offset 26368, 16B aligned)
#define LDS_BYTES (1280 + 8192 + 8192 + 512 + 8192 + 32768)

// State stored interleaved (re,im); all state accesses are b64/b128 with
// immediate ds offsets (no per-access address math).
// mode 0: state viewed as S[r=a(16), n=m*4+c(64)]    -> idx = r*64 + n
// mode 1: state viewed as S[r=m(16), n=(a,c)(64)]    -> idx = (n>>2)*64 + r*4 + (n&3)
// A-operand layout (ISA): M = lane&15; VGPR0 K = 2*(lane>>4), VGPR1 = +1 (per 4k chunk).
// B-operand assumed to mirror A's K->(VGPR,half) mapping with N = lane&15.
// D layout (ISA): VGPR v: lanes0-15 M=v, lanes16-31 M=v+8; N = lane&15.

// Left-multiply: new = U(16x16 complex) @ S-view.  64 x v_wmma_f32_16x16x4_f32.
__device__ __forceinline__ void apply16(float2* S,
                                        const float* Ure, const float* Uim,
                                        const int mode, const int lane) {
  const int hf  = lane >> 4;
  const int nlo = lane & 15;

  // Hoist the gate-matrix (A) operands: identical for all 4 output tiles.
  v2f aRe[4], aIm[4], aImN[4];
  #pragma unroll
  for (int k = 0; k < 4; ++k) {
    const int K0 = 4 * k + 2 * hf;
    aRe[k].x  = Ure[nlo * 16 + K0];  aRe[k].y  = Ure[nlo * 16 + K0 + 1];
    aIm[k].x  = Uim[nlo * 16 + K0];  aIm[k].y  = Uim[nlo * 16 + K0 + 1];
    aImN[k].x = -aIm[k].x;           aImN[k].y = -aIm[k].y;  // no f32 A-negate modifier
  }

  #pragma unroll
  for (int jt = 0; jt < 4; ++jt) {          // 4 column tiles of 16 (N = 64 total)
    const int n = jt * 16 + nlo;
    v8f accRe = {0.f,0.f,0.f,0.f,0.f,0.f,0.f,0.f};
    v8f accIm = {0.f,0.f,0.f,0.f,0.f,0.f,0.f,0.f};
    #pragma unroll
    for (int k = 0; k < 4; ++k) {           // K = 16 in chunks of 4
      const int K0 = 4 * k + 2 * hf;
      const int K1 = K0 + 1;
      const int i0 = (mode == 0) ? (K0 * 64 + n) : (((n >> 2) << 6) + K0 * 4 + (n & 3));
      const int i1 = (mode == 0) ? (K1 * 64 + n) : (((n >> 2) << 6) + K1 * 4 + (n & 3));
      const float2 z0 = S[i0];              // b64, immediate offset
      const float2 z1 = S[i1];
      v2f br, bi;                           // pairing movs dual-issue
      br.x = z0.x;  br.y = z1.x;
      bi.x = z0.y;  bi.y = z1.y;
      // newRe += Ure*Sre - Uim*Sim ; newIm += Ure*Sim + Uim*Sre
      accRe = __builtin_amdgcn_wmma_f32_16x16x4_f32(false, aRe[k],  false, br, (short)0, accRe, false, false);
      accRe = __builtin_amdgcn_wmma_f32_16x16x4_f32(false, aImN[k], false, bi, (short)0, accRe, false, false);
      accIm = __builtin_amdgcn_wmma_f32_16x16x4_f32(false, aRe[k],  false, bi, (short)0, accIm, false, false);
      accIm = __builtin_amdgcn_wmma_f32_16x16x4_f32(false, aIm[k],  false, br, (short)0, accIm, false, false);
    }
    // In-place per tile is safe: loads (above) precede stores, same column set,
    // LDS ops are in-order within a wave.
    #pragma unroll
    for (int v = 0; v < 8; ++v) {
      const int M = v + 8 * hf;
      const int o = (mode == 0) ? (M * 64 + n) : (((n >> 2) << 6) + M * 4 + (n & 3));
      S[o] = make_float2(accRe[v], accIm[v]);   // b64, immediate offset
    }
  }
}

// Qubits {8,9}: right-multiply D = S_view1 @ (I4 (x) UC^T).  The 64x64 operator is
// block-diagonal and tiles into the SAME 16x16 block-diag matrix per 16-column
// tile, so the state becomes the A operand and B is synthesized from 4 UC
// coefficients per lane (W[K][n] = (K>>2==n>>2) ? UC[n&3][K&3] : 0).
__device__ __forceinline__ void applyC(float2* S,
                                       const float* ucr, const float* uci,
                                       const int lane) {
  const int hf  = lane >> 4;
  const int nlo = lane & 15;
  const int grp = nlo >> 2, cc = nlo & 3;
  const int c0 = 2 * hf;

  // The only nonzero B values for this lane: UC[cc][c0], UC[cc][c0+1]
  const float vrx = ucr[cc * 4 + c0], vry = ucr[cc * 4 + c0 + 1];
  const float vix = uci[cc * 4 + c0], viy = uci[cc * 4 + c0 + 1];
  v2f bRe[4], bIm[4], bImN[4];
  #pragma unroll
  for (int k = 0; k < 4; ++k) {
    const bool on = (k == grp);
    bRe[k].x  = on ?  vrx : 0.f;  bRe[k].y  = on ?  vry : 0.f;
    bIm[k].x  = on ?  vix : 0.f;  bIm[k].y  = on ?  viy : 0.f;
    bImN[k].x = on ? -vix : 0.f;  bImN[k].y = on ? -viy : 0.f;
  }

  #pragma unroll
  for (int jt = 0; jt < 4; ++jt) {
    const int n = jt * 16 + nlo;                       // output column (mode-1 view)
    v8f accRe = {0.f,0.f,0.f,0.f,0.f,0.f,0.f,0.f};
    v8f accIm = {0.f,0.f,0.f,0.f,0.f,0.f,0.f,0.f};
    #pragma unroll
    for (int k = 0; k < 4; ++k) {
      // A = state tile: A[M=nlo][K] = S[idx1(nlo, jt*16+K)]; the K0,K1 elements are
      // ADJACENT (i0 even) -> one b128 load fetches both complex values.
      const int i0 = (4 * jt + k) * 64 + nlo * 4 + 2 * hf;   // idx1(nlo, 16jt+4k+2hf)
      const float4 z = *(const float4*)(S + i0);             // (re0,im0,re1,im1)
      v2f ar, ai;
      ar.x = z.x;  ar.y = z.z;
      ai.x = z.y;  ai.y = z.w;
      // D = A*B (complex): Dre += Ar*Br - Ai*Bi ; Dim += Ar*Bi + Ai*Br
      accRe = __builtin_amdgcn_wmma_f32_16x16x4_f32(false, ar, false, bRe[k],  (short)0, accRe, false, false);
      accRe = __builtin_amdgcn_wmma_f32_16x16x4_f32(false, ai, false, bImN[k], (short)0, accRe, false, false);
      accIm = __builtin_amdgcn_wmma_f32_16x16x4_f32(false, ar, false, bIm[k],  (short)0, accIm, false, false);
      accIm = __builtin_amdgcn_wmma_f32_16x16x4_f32(false, ai, false, bRe[k],  (short)0, accIm, false, false);
    }
    #pragma unroll
    for (int v = 0; v < 8; ++v) {
      const int M = v + 8 * hf;
      const int o = ((n >> 2) << 6) + M * 4 + (n & 3);       // idx1(M, n)
      S[o] = make_float2(accRe[v], accIm[v]);
    }
  }
}

extern "C" __global__ __launch_bounds__(THREADS)
void qlayer_kernel(const float* __restrict__ x,
                   const float* __restrict__ qw,
                   float* __restrict__ out) {
  extern __shared__ char smem_raw[];
  float* u2   = (float*)smem_raw;                 // [NL][NQ][row][col][re/im]
  float* UAre = u2 + NL * NQ * 8;                 // [NL][16*16] qubits 0..3 kron
  float* UAim = UAre + NL * 256;
  float* UBre = UAim + NL * 256;                  // qubits 4..7 kron
  float* UBim = UBre + NL * 256;
  float* UCre = UBim + NL * 256;                  // [NL][4*4] qubits 8..9 kron
  float* UCim = UCre + NL * 16;
  unsigned* permPk = (unsigned*)(UCim + NL * 16); // [NL][512]: two gather srcs per word
  float2* stateBase = (float2*)(permPk + NL * 512); // [WAVES][1024] interleaved (re,im)

  const int tid = threadIdx.x;

  // ---- shared table prep (per workgroup; trivial cost) ----
  // Per-qubit Rot gates: m00=ep*c, m01=-conj(em)*s, m10=em*s, m11=conj(ep)*c
  #pragma unroll 1
  for (int e = tid; e < NL * NQ; e += THREADS) {
    const int l = e / NQ, w = e % NQ;
    const float phi = qw[(l * NQ + w) * 3 + 0];
    const float th  = qw[(l * NQ + w) * 3 + 1];
    const float om  = qw[(l * NQ + w) * 3 + 2];
    const float c = cosf(0.5f * th), s = sinf(0.5f * th);
    const float ap = 0.5f * (phi + om), am = 0.5f * (phi - om);
    const float cpo = cosf(ap), spo = sinf(ap);
    const float cb = cosf(am),  sb = sinf(am);
    float* g = u2 + e * 8;
    g[0] =  cpo * c; g[1] = -spo * c;   // m00
    g[2] = -cb * s;  g[3] = -sb * s;    // m01
    g[4] =  cb * s;  g[5] = -sb * s;    // m10
    g[6] =  cpo * c; g[7] =  spo * c;   // m11
  }
  __syncthreads();

  // Kron products U0⊗U1⊗U2⊗U3 (UA) and U4⊗..⊗U7 (UB); qubit q sits at bit (3-q).
  #pragma unroll 1
  for (int e = tid; e < NL * 256; e += THREADS) {
    const int l = e >> 8, rc = e & 255, r = rc >> 4, cc = rc & 15;
    float arA = 1.f, aiA = 0.f, arB = 1.f, aiB = 0.f;
    #pragma unroll
    for (int q = 0; q < 4; ++q) {
      const int rb = (r >> (3 - q)) & 1, cbit = (cc >> (3 - q)) & 1;
      const float* gA = u2 + (((l * NQ + q) * 4) + rb * 2 + cbit) * 2;
      float nr = arA * gA[0] - aiA * gA[1];
      float ni = arA * gA[1] + aiA * gA[0];
      arA = nr; aiA = ni;
      const float* gB = u2 + (((l * NQ + 4 + q) * 4) + rb * 2 + cbit) * 2;
      nr = arB * gB[0] - aiB * gB[1];
      ni = arB * gB[1] + aiB * gB[0];
      arB = nr; aiB = ni;
    }
    UAre[e] = arA; UAim[e] = aiA;
    UBre[e] = arB; UBim[e] = aiB;
  }
  // 4x4 kron U8⊗U9
  #pragma unroll 1
  for (int e = tid; e < NL * 16; e += THREADS) {
    const int l = e >> 4, r = (e >> 2) & 3, cc = e & 3;
    const float* g8 = u2 + (((l * NQ + 8) * 4) + ((r >> 1) & 1) * 2 + ((cc >> 1) & 1)) * 2;
    const float* g9 = u2 + (((l * NQ + 9) * 4) + (r & 1) * 2 + (cc & 1)) * 2;
    UCre[e] = g8[0] * g9[0] - g8[1] * g9[1];
    UCim[e] = g8[0] * g9[1] + g8[1] * g9[0];
  }
  // CNOT-ring layer as a gather permutation: new[i] = old[c0(c1(...c9(i)))]
  // (CNOT is an involution; qubit w lives at flat bit 9-w.)
  // Packed: word (l, lane + 32*h) holds srcs for destinations i = lane+64h (lo 16)
  // and i = lane+64h+32 (hi 16).
  #pragma unroll 1
  for (int e = tid; e < NL * 512; e += THREADS) {
    const int l = e >> 9, p = e & 511;
    const int ln = p & 31, h = p >> 5;
    const int rr = (l % (NQ - 1)) + 1;
    unsigned pk = 0;
    #pragma unroll
    for (int half = 0; half < 2; ++half) {
      int j = ln + 64 * h + 32 * half;
      for (int w = NQ - 1; w >= 0; --w) {
        const int cpos = 9 - w;
        const int tpos = 9 - ((w + rr) % NQ);
        j ^= ((j >> cpos) & 1) << tpos;
      }
      pk |= ((unsigned)j) << (16 * half);
    }
    permPk[e] = pk;
  }
  __syncthreads();

  // ---- per-wave simulation: one batch element per wave32 ----
  const int wave = tid >> 5, lane = tid & 31;
  const int b = blockIdx.x * WAVES_PER_BLOCK + wave;     // grid sized exactly: b < 2048
  float2* S = stateBase + wave * NSTATES;

  // Angle embedding collapses to a real product state: amp[i] = prod (bit? sin : cos)
  float cw[NQ], sw[NQ];
  #pragma unroll
  for (int w = 0; w < NQ; ++w) {
    const float ang = 1.5707963267948966f * x[b * NQ + w];  // pi*x/2
    cw[w] = cosf(ang);
    sw[w] = sinf(ang);
  }
  for (int t = 0; t < 32; ++t) {
    const int i = lane + 32 * t;
    float a = 1.f;
    #pragma unroll
    for (int w = 0; w < NQ; ++w) a *= ((i >> (9 - w)) & 1) ? sw[w] : cw[w];
    S[i] = make_float2(a, 0.f);
  }

  #pragma unroll 1
  for (int l = 0; l < NL; ++l) {
    apply16(S, UAre + l * 256, UAim + l * 256, 0, lane);  // qubits 0..3 (WMMA)
    apply16(S, UBre + l * 256, UBim + l * 256, 1, lane);  // qubits 4..7 (WMMA)
    applyC (S, UCre + l * 16,  UCim + l * 16,  lane);     // qubits 8..9 (WMMA)

    // CNOT ring: stage the full permutation through registers (64 VGPRs);
    // all gathers precede all scatters in program order; same-wave LDS is in-order.
    const unsigned* pl = permPk + l * 512;
    float2 tv[32];
    #pragma unroll
    for (int h = 0; h < 16; ++h) {
      const unsigned pk = pl[lane + 32 * h];
      tv[2 * h]     = S[pk & 0xFFFFu];
      tv[2 * h + 1] = S[pk >> 16];
    }
    #pragma unroll
    for (int t = 0; t < 32; ++t) S[lane + 32 * t] = tv[t];
  }

  // Expectations <Z_i> = sum_b |amp|^2 * (1 - 2*bit_i)
  float acc[NQ];
  #pragma unroll
  for (int w = 0; w < NQ; ++w) acc[w] = 0.f;
  for (int t = 0; t < 32; ++t) {
    const int i = lane + 32 * t;
    const float2 z = S[i];
    const float p = z.x * z.x + z.y * z.y;
    #pragma unroll
    for (int w = 0; w < NQ; ++w) acc[w] += ((i >> (9 - w)) & 1) ? -p : p;
  }
  #pragma unroll
  for (int w = 0; w < NQ; ++w) {
    float v = acc[w];
    #pragma unroll
    for (int off = 16; off > 0; off >>= 1) v += __shfl_xor(v, off, 32);
    if (lane == 0) out[b * NQ + w] = v;
  }
}

extern "C" void kernel_launch(void* const* d_in, const int* in_sizes, int n_in,
                              void* d_out, int out_size, void* d_ws, size_t ws_size,
                              hipStream_t stream) {
  (void)in_sizes; (void)n_in; (void)out_size; (void)d_ws; (void)ws_size;
  const float* x  = (const float*)d_in[0];   // (2048, 10) f32
  const float* qw = (const float*)d_in[1];   // (4, 10, 3) f32
  float* out = (float*)d_out;                // (2048, 10) f32
  const int batch = 2048;
  dim3 grid(batch / WAVES_PER_BLOCK);        // 512 blocks
  dim3 block(THREADS);                       // 128 threads = 4 wave32
  qlayer_kernel<<<grid, block, LDS_BYTES, stream>>>(x, qw, out);
}